// VectorQuantizer_60507499266847
// MI455X (gfx1250) — compile-verified
//
#include <hip/hip_runtime.h>

typedef __attribute__((ext_vector_type(2))) float v2f;
typedef __attribute__((ext_vector_type(4))) float v4f;
typedef __attribute__((ext_vector_type(8))) float v8f;

#define DIM        64
#define KCODES     512
#define HW         4096     // H*W
#define CH_STRIDE  4096     // elements between channel planes
#define B_STRIDE   262144   // 64 * 4096
#define PTS_PER_WG 128
#define PAIR_PITCH 1056     // 512 codes * 2 dwords + 32 pad => pitch % 64 == 32:
                            // halves of the wave land on disjoint bank ranges
#define TOTAL_ELEMS 8388608 // 32*64*64*64

// LDS codebook layout: s_cbP[pairRow][code][2] with pairRow = dim/2.
// A lane's B-fragment (dims 2r, 2r+1 at one code) is a single aligned b64.

__global__ __launch_bounds__(256) void vq_main(const float* __restrict__ z,
                                               const float* __restrict__ cb,
                                               float* __restrict__ out,
                                               float* __restrict__ partial) {
  __shared__ float s_cbP[32 * PAIR_PITCH];   // 135168 B
  __shared__ float s_enorm[KCODES];
  __shared__ int   s_best[PTS_PER_WG];
  __shared__ float s_red[8];

  const int tid  = threadIdx.x;
  const int wg   = blockIdx.x;
  const int n0   = wg * PTS_PER_WG;
  const int b    = n0 >> 12;                 // n0 / 4096
  const int rem  = n0 & (HW - 1);
  const long zbase = (long)b * B_STRIDE + rem;

  // ---- stage codebook into LDS (pair-interleaved), coalesced float4 reads
  const v4f* cb4 = (const v4f*)cb;
  #pragma unroll
  for (int i = 0; i < 32; ++i) {
    int idx4 = tid + i * 256;                // 8192 float4 total
    v4f v = cb4[idx4];
    int j  = idx4 >> 4;                      // code index
    int d0 = (idx4 & 15) << 2;               // dim base: 0,4,...,60
    int pr = d0 >> 1;                        // pair row: even
    v2f lo; lo.x = v.x; lo.y = v.y;
    v2f hi2; hi2.x = v.z; hi2.y = v.w;
    *(v2f*)&s_cbP[(pr + 0) * PAIR_PITCH + 2 * j] = lo;   // 8B-aligned ds_store_b64
    *(v2f*)&s_cbP[(pr + 1) * PAIR_PITCH + 2 * j] = hi2;
  }
  __syncthreads();

  // ---- per-code squared norms
  for (int j = tid; j < KCODES; j += 256) {
    float s = 0.f;
    #pragma unroll
    for (int pr = 0; pr < 32; ++pr) {
      v2f e = *(const v2f*)&s_cbP[pr * PAIR_PITCH + 2 * j];
      s += e.x * e.x + e.y * e.y;
    }
    s_enorm[j] = s;
  }
  __syncthreads();

  // ---- per-wave: 16 points, fp32 WMMA distances vs all 512 codes
  const int lane  = tid & 31;
  const int wv    = tid >> 5;
  const int laneM = lane & 15;
  const int hi    = lane >> 4;               // 0: K=4i,4i+1  1: K=4i+2,4i+3
  const int p0    = wv * 16;

  // A fragments: full D=64 for this wave's 16 points (reused for all 32 tiles)
  float a[32];
  #pragma unroll
  for (int i = 0; i < 16; ++i) {
    int c0 = 4 * i + 2 * hi;
    a[2 * i]     = z[zbase + (long)c0       * CH_STRIDE + p0 + laneM];
    a[2 * i + 1] = z[zbase + (long)(c0 + 1) * CH_STRIDE + p0 + laneM];
  }

  float bestd[8];
  int   besti[8];
  #pragma unroll
  for (int r = 0; r < 8; ++r) { bestd[r] = 3.4e38f; besti[r] = 0x7fffffff; }

  for (int t = 0; t < 32; ++t) {
    const int j0 = t * 16;
    const int jcol = 2 * (j0 + laneM);       // dword offset of this lane's code
    v8f acc0 = {}; v8f acc1 = {};
    #pragma unroll
    for (int i = 0; i < 16; i += 2) {
      {
        // pair row for this lane's dims at k-step i: 2*i + hi
        v2f av; av.x = a[2 * i]; av.y = a[2 * i + 1];
        v2f bv = *(const v2f*)&s_cbP[(2 * i + hi) * PAIR_PITCH + jcol];
        acc0 = __builtin_amdgcn_wmma_f32_16x16x4_f32(
            false, av, false, bv, (short)0, acc0, false, false);
      }
      {
        v2f av; av.x = a[2 * (i + 1)]; av.y = a[2 * (i + 1) + 1];
        v2f bv = *(const v2f*)&s_cbP[(2 * (i + 1) + hi) * PAIR_PITCH + jcol];
        acc1 = __builtin_amdgcn_wmma_f32_16x16x4_f32(
            false, av, false, bv, (short)0, acc1, false, false);
      }
    }
    const float en  = s_enorm[j0 + laneM];
    const int   myj = j0 + laneM;
    #pragma unroll
    for (int r = 0; r < 8; ++r) {
      float d = en - 2.0f * (acc0[r] + acc1[r]);  // ||z||^2 constant per row
      if (d < bestd[r]) { bestd[r] = d; besti[r] = myj; }
    }
  }

  // ---- argmin across the 16 codes held in each 16-lane half (C layout split)
  #pragma unroll
  for (int r = 0; r < 8; ++r) {
    float dv = bestd[r]; int iv = besti[r];
    #pragma unroll
    for (int off = 8; off >= 1; off >>= 1) {
      float od = __shfl_xor(dv, off, 32);
      int   oi = __shfl_xor(iv, off, 32);
      if (od < dv || (od == dv && oi < iv)) { dv = od; iv = oi; }
    }
    besti[r] = iv;
  }
  if (lane == 0) {
    #pragma unroll
    for (int r = 0; r < 8; ++r) s_best[p0 + r] = besti[r];       // M = r
  } else if (lane == 16) {
    #pragma unroll
    for (int r = 0; r < 8; ++r) s_best[p0 + 8 + r] = besti[r];   // M = r + 8
  }
  __syncthreads();

  // ---- gather + coalesced store + MSE (q and loss in one pass)
  float sq = 0.f;
  #pragma unroll
  for (int i = 0; i < 32; ++i) {
    int idx = tid + i * 256;                 // 64 dims x 128 points
    int c = idx >> 7;
    int p = idx & 127;
    int bi = s_best[p];
    float q  = s_cbP[(c >> 1) * PAIR_PITCH + 2 * bi + (c & 1)];
    long ga  = zbase + (long)c * CH_STRIDE + p;
    float zv = z[ga];
    out[ga] = q;                             // straight-through output == q
    float df = q - zv;
    sq += df * df;
  }
  #pragma unroll
  for (int off = 16; off >= 1; off >>= 1) sq += __shfl_xor(sq, off, 32);
  if (lane == 0) s_red[wv] = sq;
  __syncthreads();
  if (tid == 0) {
    float tot = 0.f;
    #pragma unroll
    for (int i = 0; i < 8; ++i) tot += s_red[i];
    partial[wg] = tot;                       // deterministic: no atomics
  }
}

__global__ __launch_bounds__(256) void vq_reduce(const float* __restrict__ partial,
                                                 float* __restrict__ out) {
  __shared__ float s[256];
  int tid = threadIdx.x;
  float v = 0.f;
  #pragma unroll
  for (int i = 0; i < 4; ++i) v += partial[tid + i * 256];   // 1024 partials
  s[tid] = v;
  __syncthreads();
  for (int stride = 128; stride >= 1; stride >>= 1) {
    if (tid < stride) s[tid] += s[tid + stride];
    __syncthreads();
  }
  if (tid == 0) {
    // loss = q_latent + 0.25*e_latent = 1.25 * mean((q - z)^2)
    out[TOTAL_ELEMS] = 1.25f * s[0] / (float)TOTAL_ELEMS;
  }
}

extern "C" void kernel_launch(void* const* d_in, const int* in_sizes, int n_in,
                              void* d_out, int out_size, void* d_ws, size_t ws_size,
                              hipStream_t stream) {
  (void)in_sizes; (void)n_in; (void)out_size; (void)ws_size;
  const float* z  = (const float*)d_in[0];
  const float* cb = (const float*)d_in[1];
  float* out      = (float*)d_out;
  float* partial  = (float*)d_ws;            // 1024 floats of scratch

  vq_main  <<<1024, 256, 0, stream>>>(z, cb, out, partial);
  vq_reduce<<<1,    256, 0, stream>>>(partial, out);
}